// ExperimentalTensorProduct_41369124995687
// MI455X (gfx1250) — compile-verified
//
#include <hip/hip_runtime.h>

typedef float v2f __attribute__((ext_vector_type(2)));
typedef float v8f __attribute__((ext_vector_type(8)));

// ---- constants -------------------------------------------------------------
#define C_1_SQRT3   0.5773502691896258f   // 1/sqrt(3)
#define C_1_SQRT5   0.4472135954999579f   // 1/sqrt(5)
#define C_S2        0.7071067811865476f   // 1/sqrt(2)
#define C_S6        0.4082482904638630f   // 1/sqrt(6)
#define C_35        0.7745966692414834f   // sqrt(3/5)
#define C_A         0.5345224838248488f   // sqrt(2/7)
#define C_B         0.4629100498862757f   // sqrt(3/14)
#define C_INV96     0.1020620726159657f   // 1/sqrt(96)
#define C_INV128    0.0883883476483184f   // 1/sqrt(128)

__launch_bounds__(32)
__global__ void etp_kernel(const float* __restrict__ x,
                           const float* __restrict__ y,
                           const float* __restrict__ w,
                           float* __restrict__ out)
{
    // z rows padded with zeros so WMMA B-tiles (16 wide) read exact zeros.
    __shared__ float z0[96];        // 0e: 3 segs * 32
    __shared__ float z1[128 * 4];   // 1o: 4 segs * 32 rows, 3 vals + 1 zero
    __shared__ float z2[128 * 8];   // 2e: 4 segs * 32 rows, 5 vals + 3 zeros
    __shared__ float scr[32 * 16];  // D-tile staging

    const int b    = blockIdx.x;
    const int lane = threadIdx.x;

    const float* xp = x + (size_t)b * 288;
    const float* yp = y + (size_t)b * 9;     // wave-uniform -> s_load path
    const float* wp = w + (size_t)b * 11264;
    float*       op = out + (size_t)b * 288;

    // ---- per-channel inputs (lane u handles channel u of all irreps) ------
    const float x0u = xp[lane];
    float a1v[3], a2v[5];
    a1v[0] = xp[32 + lane * 3 + 0];
    a1v[1] = xp[32 + lane * 3 + 1];
    a1v[2] = xp[32 + lane * 3 + 2];
    a2v[0] = xp[128 + lane * 5 + 0];
    a2v[1] = xp[128 + lane * 5 + 1];
    a2v[2] = xp[128 + lane * 5 + 2];
    a2v[3] = xp[128 + lane * 5 + 3];
    a2v[4] = xp[128 + lane * 5 + 4];

    const float yy0 = yp[0];
    const float Y1[3] = { yp[1], yp[2], yp[3] };
    const float Y2[5] = { yp[4], yp[5], yp[6], yp[7], yp[8] };

    // ---- tensor product: 11 parity-matched CG paths -----------------------
    // 0e segments: (0,0,0), (1,1,0), (2,2,0)
    z0[lane]      = x0u * yy0;
    z0[32 + lane] = C_1_SQRT3 * (a1v[0]*Y1[0] + a1v[1]*Y1[1] + a1v[2]*Y1[2]);
    z0[64 + lane] = C_1_SQRT5 * (a2v[0]*Y2[0] + a2v[1]*Y2[1] + a2v[2]*Y2[2]
                               + a2v[3]*Y2[3] + a2v[4]*Y2[4]);

    // (1,2,1)/(2,1,1) map: r_j = sqrt(3/5) * sum_ik (E_k)_{ij} v_i w_k
    auto map121 = [](const float v[3], const float w5[5], float r[3]) {
        r[0] = C_35 * ((v[2]*w5[0] + v[1]*w5[1]) * C_S2
                       - v[0]*w5[2] * C_S6 - v[0]*w5[4] * C_S2);
        r[1] = C_35 * (v[0]*w5[1] * C_S2 + 2.f*v[1]*w5[2] * C_S6
                       + v[2]*w5[3] * C_S2);
        r[2] = C_35 * (v[0]*w5[0] * C_S2 - v[2]*w5[2] * C_S6
                       + (v[1]*w5[3] + v[2]*w5[4]) * C_S2);
    };

    // 1o segments: (0,1,1), (1,0,1), (1,2,1), (2,1,1)
    *(float4*)&z1[lane * 4]        = make_float4(x0u*Y1[0], x0u*Y1[1], x0u*Y1[2], 0.f);
    *(float4*)&z1[(32 + lane) * 4] = make_float4(a1v[0]*yy0, a1v[1]*yy0, a1v[2]*yy0, 0.f);
    { float r[3]; map121(a1v, Y2, r);
      *(float4*)&z1[(64 + lane) * 4] = make_float4(r[0], r[1], r[2], 0.f); }
    { float r[3]; map121(Y1, a2v, r);
      *(float4*)&z1[(96 + lane) * 4] = make_float4(r[0], r[1], r[2], 0.f); }

    // 2e segments: (0,2,2), (1,1,2), (2,0,2), (2,2,2)
    *(float4*)&z2[lane * 8]     = make_float4(x0u*Y2[0], x0u*Y2[1], x0u*Y2[2], x0u*Y2[3]);
    *(float4*)&z2[lane * 8 + 4] = make_float4(x0u*Y2[4], 0.f, 0.f, 0.f);
    {   // (1,1,2): symmetric traceless basis, l=1 order (y,z,x)
        const float q0 = (a1v[2]*Y1[0] + a1v[0]*Y1[2]) * C_S2;
        const float q1 = (a1v[0]*Y1[1] + a1v[1]*Y1[0]) * C_S2;
        const float q2 = (2.f*a1v[1]*Y1[1] - a1v[2]*Y1[2] - a1v[0]*Y1[0]) * C_S6;
        const float q3 = (a1v[1]*Y1[2] + a1v[2]*Y1[1]) * C_S2;
        const float q4 = (a1v[2]*Y1[2] - a1v[0]*Y1[0]) * C_S2;
        *(float4*)&z2[(32 + lane) * 8]     = make_float4(q0, q1, q2, q3);
        *(float4*)&z2[(32 + lane) * 8 + 4] = make_float4(q4, 0.f, 0.f, 0.f);
    }
    *(float4*)&z2[(64 + lane) * 8]     = make_float4(a2v[0]*yy0, a2v[1]*yy0, a2v[2]*yy0, a2v[3]*yy0);
    *(float4*)&z2[(64 + lane) * 8 + 4] = make_float4(a2v[4]*yy0, 0.f, 0.f, 0.f);
    {   // (2,2,2): fully symmetric invariant
        const float X0=a2v[0], X1=a2v[1], X2=a2v[2], X3=a2v[3], X4=a2v[4];
        const float W0=Y2[0],  W1=Y2[1],  W2=Y2[2],  W3=Y2[3],  W4=Y2[4];
        const float p0 = -C_A*(X0*W2 + X2*W0) + C_B*(X1*W3 + X3*W1);
        const float p1 = 0.5f*C_A*(X1*W2 + X2*W1) - C_B*(X1*W4 + X4*W1) + C_B*(X0*W3 + X3*W0);
        const float p2 = C_A*(-X0*W0 + 0.5f*X1*W1 + X2*W2 + 0.5f*X3*W3 - X4*W4);
        const float p3 = 0.5f*C_A*(X3*W2 + X2*W3) + C_B*(X3*W4 + X4*W3) + C_B*(X0*W1 + X1*W0);
        const float p4 = -C_A*(X4*W2 + X2*W4) + C_B*(X3*W3 - X1*W1);
        *(float4*)&z2[(96 + lane) * 8]     = make_float4(p0, p1, p2, p3);
        *(float4*)&z2[(96 + lane) * 8 + 4] = make_float4(p4, 0.f, 0.f, 0.f);
    }
    __syncthreads();

    // ---- 0e linear: out[o] = 1/sqrt(96) * sum_u w0[u,o] * z0[u] -----------
    {
        float acc0 = 0.f;
        for (int u = 0; u < 96; u += 4) {
            const float4 zz = *(const float4*)&z0[u];           // LDS broadcast
            const float* wr = wp + u * 32 + lane;               // coalesced rows
            acc0 = fmaf(__builtin_nontemporal_load(wr),      zz.x, acc0);
            acc0 = fmaf(__builtin_nontemporal_load(wr + 32), zz.y, acc0);
            acc0 = fmaf(__builtin_nontemporal_load(wr + 64), zz.z, acc0);
            acc0 = fmaf(__builtin_nontemporal_load(wr + 96), zz.w, acc0);
        }
        op[lane] = acc0 * C_INV96;
    }

    const int lo   = lane & 15;
    const int hi   = lane >> 4;          // 0 or 1
    const int aoff = hi * 64 + lo;       // A-tile: lanes>=16 hold K+2 rows

    // ---- 1o linear via v_wmma_f32_16x16x4_f32 -----------------------------
    {
        const float* w1p = wp + 3072;
        const int n3 = lo < 3 ? lo : 3;                         // col>=3 -> zero slot
        const char* bbase = (const char*)z1 + hi * 32 + n3 * 4; // +2 rows for hi lanes
        v8f accA = {0.f,0.f,0.f,0.f,0.f,0.f,0.f,0.f};
        v8f accB = {0.f,0.f,0.f,0.f,0.f,0.f,0.f,0.f};
        for (int u0 = 0; u0 < 128; u0 += 4) {
            const float* wr = w1p + u0 * 32 + aoff;
            v2f A0, A1, Bv;
            A0.x = __builtin_nontemporal_load(wr);        // K=0 / K=2
            A0.y = __builtin_nontemporal_load(wr + 32);   // K=1 / K=3
            A1.x = __builtin_nontemporal_load(wr + 16);   // outputs 16..31
            A1.y = __builtin_nontemporal_load(wr + 48);
            const float* bp = (const float*)(bbase + u0 * 16);
            Bv.x = bp[0];                                  // row u0 (+2 for hi)
            Bv.y = bp[4];                                  // row u0+1 (+2)
            accA = __builtin_amdgcn_wmma_f32_16x16x4_f32(false, A0, false, Bv,
                                                         (short)0, accA, false, false);
            accB = __builtin_amdgcn_wmma_f32_16x16x4_f32(false, A1, false, Bv,
                                                         (short)0, accB, false, false);
        }
#pragma unroll
        for (int r = 0; r < 8; ++r) scr[(hi * 8 + r) * 16 + lo]        = accA[r] * C_INV128;
#pragma unroll
        for (int r = 0; r < 8; ++r) scr[(16 + hi * 8 + r) * 16 + lo]   = accB[r] * C_INV128;
    }
    __syncthreads();
#pragma unroll
    for (int j = 0; j < 3; ++j) {
        const int idx = lane + 32 * j;
        const int o = idx / 3, k = idx - o * 3;
        op[32 + idx] = scr[o * 16 + k];
    }
    __syncthreads();

    // ---- 2e linear via v_wmma_f32_16x16x4_f32 -----------------------------
    {
        const float* w2p = wp + 7168;
        const int n5 = lo < 5 ? lo : 5;
        const char* bbase = (const char*)z2 + hi * 64 + n5 * 4; // 32B row stride
        v8f accA = {0.f,0.f,0.f,0.f,0.f,0.f,0.f,0.f};
        v8f accB = {0.f,0.f,0.f,0.f,0.f,0.f,0.f,0.f};
        for (int u0 = 0; u0 < 128; u0 += 4) {
            const float* wr = w2p + u0 * 32 + aoff;
            v2f A0, A1, Bv;
            A0.x = __builtin_nontemporal_load(wr);
            A0.y = __builtin_nontemporal_load(wr + 32);
            A1.x = __builtin_nontemporal_load(wr + 16);
            A1.y = __builtin_nontemporal_load(wr + 48);
            const float* bp = (const float*)(bbase + u0 * 32);
            Bv.x = bp[0];
            Bv.y = bp[8];
            accA = __builtin_amdgcn_wmma_f32_16x16x4_f32(false, A0, false, Bv,
                                                         (short)0, accA, false, false);
            accB = __builtin_amdgcn_wmma_f32_16x16x4_f32(false, A1, false, Bv,
                                                         (short)0, accB, false, false);
        }
#pragma unroll
        for (int r = 0; r < 8; ++r) scr[(hi * 8 + r) * 16 + lo]        = accA[r] * C_INV128;
#pragma unroll
        for (int r = 0; r < 8; ++r) scr[(16 + hi * 8 + r) * 16 + lo]   = accB[r] * C_INV128;
    }
    __syncthreads();
#pragma unroll
    for (int j = 0; j < 5; ++j) {
        const int idx = lane + 32 * j;
        const int o = idx / 5, k = idx - o * 5;
        op[128 + idx] = scr[o * 16 + k];
    }
}

extern "C" void kernel_launch(void* const* d_in, const int* in_sizes, int n_in,
                              void* d_out, int out_size, void* d_ws, size_t ws_size,
                              hipStream_t stream) {
    const float* x = (const float*)d_in[0];
    const float* y = (const float*)d_in[1];
    const float* w = (const float*)d_in[2];
    float* out     = (float*)d_out;
    const int B = in_sizes[1] / 9;   // 8192
    etp_kernel<<<dim3(B), dim3(32), 0, stream>>>(x, y, w, out);
}